// ByteToSpan_27023934226954
// MI455X (gfx1250) — compile-verified
//
#include <hip/hip_runtime.h>
#include <math.h>

// Problem constants (match reference)
#define Bc  64
#define Sc  256
#define Tc  32
#define Ec  256
#define Hc  1024
#define Vc  512
#define OEc 256
#define NG  4096   // 4*H (gate width)

#define SLAB_K 64  // K-rows of W staged per LDS slab (multiple of 4; divides K1,K2)

typedef float v2f __attribute__((ext_vector_type(2)));
typedef float v8f __attribute__((ext_vector_type(8)));

__device__ __forceinline__ float sigm(float x) { return 1.0f / (1.0f + __expf(-x)); }

__global__ void zero_kernel(float* __restrict__ p, int n) {
  int i = blockIdx.x * blockDim.x + threadIdx.x;
  if (i < n) p[i] = 0.0f;
}

// LDS slab layout: element (k, n) of the 64x16 W slab lives at float offset
//   (k>>1)*32 + n*2 + (k&1)
// so that a lane's B-fragment (Bf.x = W[4c+2h][n], Bf.y = W[4c+2h+1][n]) is one
// contiguous b64 at offset ((c*2+h)*32 + n*2). Conflict-free: half-wave 0 spans
// banks 0-31, half-wave 1 spans banks 32-63.
__device__ __forceinline__ int slab_off(int k, int n) {
  return ((k >> 1) << 5) + (n << 1) + (k & 1);
}

// Z[B,N] = concat(A1, A2) @ W + bias.
// A1 rows: embedding gather (idx != nullptr) or direct rows of a1 [B,K1].
// A2: recurrent h state [B,K2]. K1, K2 multiples of SLAB_K.
// Block = 4 waves; wave w computes M-tile [16w,16w+16) x N-tile [16*blockIdx.x, +16).
// The W slab is shared across all 4 M-tiles via LDS (4x traffic cut, M=64 total).
__global__ void lstm_gemm(const float* __restrict__ emb,
                          const int*   __restrict__ idx, int idxStride,
                          const float* __restrict__ a1, int K1,
                          const float* __restrict__ a2, int K2,
                          const float* __restrict__ W,     // [(K1+K2), N]
                          const float* __restrict__ bias,  // [N]
                          float*       __restrict__ Z,     // [B, N]
                          int N)
{
  __shared__ float sB[SLAB_K * 16];   // 4 KB

  const int tid  = threadIdx.x;       // 0..127
  const int lane = tid & 31;
  const int wave = tid >> 5;          // 0..3 -> M-tile
  const int half = lane >> 4;         // K sub-pair within chunk
  const int lm   = lane & 15;
  const int n0   = blockIdx.x << 4;
  const int m0   = wave << 4;
  const int m    = m0 + lm;
  const int n    = n0 + lm;

  const float* rowA1 = idx ? (emb + (size_t)idx[(size_t)m * idxStride] * K1)
                           : (a1  + (size_t)m * K1);
  const float* rowA2 = a2 + (size_t)m * K2;

  v8f acc = {};
  const int K = K1 + K2;

  for (int ks = 0; ks < K; ks += SLAB_K) {
    // ---- cooperative stage: W[ks..ks+63][n0..n0+15] -> sB (fragment layout)
    #pragma unroll
    for (int it = 0; it < (SLAB_K * 16) / 128; ++it) {   // 8 iters
      int i  = it * 128 + tid;
      int kk = i >> 4;              // 0..63
      int nn = i & 15;
      sB[slab_off(kk, nn)] = W[(size_t)(ks + kk) * N + n0 + nn];
    }
    __syncthreads();

    // ---- consume: 16 K-chunks of 4
    const float* rA = (ks < K1) ? (rowA1 + ks) : (rowA2 + (ks - K1));
    #pragma unroll
    for (int c = 0; c < SLAB_K / 4; ++c) {
      v2f A  = *(const v2f*)(rA + c * 4 + half * 2);
      v2f Bf = *(const v2f*)(&sB[((c * 2 + half) << 5) + (lm << 1)]);
      acc = __builtin_amdgcn_wmma_f32_16x16x4_f32(false, A, false, Bf,
                                                  (short)0, acc, false, false);
    }
    __syncthreads();                // protect sB before next slab overwrite
  }

  #pragma unroll
  for (int r = 0; r < 8; ++r) {
    int mo = m0 + half * 8 + r;     // C/D layout: rows r / r+8 per half-wave
    Z[(size_t)mo * N + n] = acc[r] + bias[n];
  }
}

// Pointwise LSTM cell: z = [i|j|f|o] (each H wide), forget_bias = 1.0.
// lens != nullptr -> freeze state past sequence length, but hfeed always gets raw h2
// (dynamic_rnn semantics: next layer consumes unmasked h2).
__global__ void lstm_cell(const float* __restrict__ z,
                          float* __restrict__ c, float* __restrict__ h,
                          float* __restrict__ hfeed,
                          const int* __restrict__ lens, int t)
{
  int i  = blockIdx.x * blockDim.x + threadIdx.x;   // 0 .. B*H-1
  int b  = i >> 10;                                  // H = 1024
  int hh = i & (Hc - 1);
  const float* zb = z + (size_t)b * NG;
  float zi = zb[hh];
  float zj = zb[Hc + hh];
  float zf = zb[2 * Hc + hh];
  float zo = zb[3 * Hc + hh];
  float cold = c[i];
  float c2 = sigm(zf + 1.0f) * cold + sigm(zi) * tanhf(zj);
  float h2 = tanhf(c2) * sigm(zo);
  bool upd = (lens == nullptr) || (t < lens[b]);
  if (upd) { c[i] = c2; h[i] = h2; }
  if (hfeed) hfeed[i] = h2;
}

// logits[:, t, :] = keep ? h1 @ proj_W : 0   (keep = t < target_lens[b])
// Same LDS-slab WMMA structure; K = H, N = V.
__global__ void proj_gemm(const float* __restrict__ hvec,   // [B, H]
                          const float* __restrict__ Wp,     // [H, V]
                          const int*   __restrict__ tlens,  // [B]
                          int t,
                          float*       __restrict__ out)    // [B, T, V]
{
  __shared__ float sB[SLAB_K * 16];

  const int tid  = threadIdx.x;
  const int lane = tid & 31;
  const int wave = tid >> 5;
  const int half = lane >> 4;
  const int lm   = lane & 15;
  const int n0   = blockIdx.x << 4;
  const int m0   = wave << 4;
  const int m    = m0 + lm;
  const int n    = n0 + lm;

  const float* rowA = hvec + (size_t)m * Hc;
  v8f acc = {};

  for (int ks = 0; ks < Hc; ks += SLAB_K) {
    #pragma unroll
    for (int it = 0; it < (SLAB_K * 16) / 128; ++it) {
      int i  = it * 128 + tid;
      int kk = i >> 4;
      int nn = i & 15;
      sB[slab_off(kk, nn)] = Wp[(size_t)(ks + kk) * Vc + n0 + nn];
    }
    __syncthreads();

    #pragma unroll
    for (int c = 0; c < SLAB_K / 4; ++c) {
      v2f A  = *(const v2f*)(rowA + ks + c * 4 + half * 2);
      v2f Bf = *(const v2f*)(&sB[((c * 2 + half) << 5) + (lm << 1)]);
      acc = __builtin_amdgcn_wmma_f32_16x16x4_f32(false, A, false, Bf,
                                                  (short)0, acc, false, false);
    }
    __syncthreads();
  }

  #pragma unroll
  for (int r = 0; r < 8; ++r) {
    int mo = m0 + half * 8 + r;
    float val = (t < tlens[mo]) ? acc[r] : 0.0f;
    out[((size_t)mo * Tc + t) * Vc + n] = val;
  }
}

extern "C" void kernel_launch(void* const* d_in, const int* in_sizes, int n_in,
                              void* d_out, int out_size, void* d_ws, size_t ws_size,
                              hipStream_t stream)
{
  const int*   input_symbols = (const int*)  d_in[0];   // [B,S]
  const int*   lengths       = (const int*)  d_in[1];   // [B]
  const int*   target_inputs = (const int*)  d_in[2];   // [B,T]
  const int*   target_lens   = (const int*)  d_in[3];   // [B]
  const float* byte_emb      = (const float*)d_in[4];   // [257,E]
  const float* enc_W0        = (const float*)d_in[5];   // [E+H, 4H]
  const float* enc_b0        = (const float*)d_in[6];   // [4H]
  const float* enc_W1        = (const float*)d_in[7];   // [2H, 4H]
  const float* enc_b1        = (const float*)d_in[8];
  const float* out_emb       = (const float*)d_in[9];   // [V, OE]
  const float* dec_W0        = (const float*)d_in[10];  // [OE+H, 4H]
  const float* dec_b0        = (const float*)d_in[11];
  const float* dec_W1        = (const float*)d_in[12];  // [2H, 4H]
  const float* dec_b1        = (const float*)d_in[13];
  const float* proj_W        = (const float*)d_in[14];  // [H, V]
  float* out = (float*)d_out;

  float* ws = (float*)d_ws;
  const size_t BH = (size_t)Bc * Hc;
  float* c0    = ws;
  float* h0    = c0 + BH;
  float* c1    = h0 + BH;
  float* h1    = c1 + BH;
  float* hfeed = h1 + BH;
  float* zbuf  = hfeed + BH;      // [B, 4H] gate buffer

  // Zero all LSTM state each call (workspace is poisoned, not re-zeroed by harness).
  {
    int n = (int)(4 * BH);
    zero_kernel<<<(n + 255) / 256, 256, 0, stream>>>(c0, n);
  }

  dim3 gGrid(NG / 16);            // 256 N-strips; block covers full M=64
  dim3 pGrid(Vc / 16);            // 32 N-strips
  dim3 cGrid((Bc * Hc) / 256);

  // ---- Encoder: 256 sequential steps, state frozen past `lengths`
  for (int t = 0; t < Sc; ++t) {
    lstm_gemm<<<gGrid, 128, 0, stream>>>(byte_emb, input_symbols + t, Sc,
                                         nullptr, Ec, h0, Hc,
                                         enc_W0, enc_b0, zbuf, NG);
    lstm_cell<<<cGrid, 256, 0, stream>>>(zbuf, c0, h0, hfeed, lengths, t);
    lstm_gemm<<<gGrid, 128, 0, stream>>>(nullptr, nullptr, 0,
                                         hfeed, Hc, h1, Hc,
                                         enc_W1, enc_b1, zbuf, NG);
    lstm_cell<<<cGrid, 256, 0, stream>>>(zbuf, c1, h1, nullptr, lengths, t);
  }

  // ---- Decoder: teacher-forced 32 steps from encoder state; masked logits out
  for (int t = 0; t < Tc; ++t) {
    lstm_gemm<<<gGrid, 128, 0, stream>>>(out_emb, target_inputs + t, Tc,
                                         nullptr, OEc, h0, Hc,
                                         dec_W0, dec_b0, zbuf, NG);
    lstm_cell<<<cGrid, 256, 0, stream>>>(zbuf, c0, h0, hfeed, nullptr, t);
    lstm_gemm<<<gGrid, 128, 0, stream>>>(nullptr, nullptr, 0,
                                         hfeed, Hc, h1, Hc,
                                         dec_W1, dec_b1, zbuf, NG);
    lstm_cell<<<cGrid, 256, 0, stream>>>(zbuf, c1, h1, nullptr, nullptr, t);
    proj_gemm<<<pGrid, 128, 0, stream>>>(h1, proj_W, target_lens, t, out);
  }
}